// GraphAttentionLayer_89507118449099
// MI455X (gfx1250) — compile-verified
//
#include <hip/hip_runtime.h>

typedef __attribute__((ext_vector_type(2))) float v2f;
typedef __attribute__((ext_vector_type(8))) float v8f;

#define NROWS 8192
#define FIN   512
#define FOUT  256
#define ALPHA 0.2f

// ---------------------------------------------------------------------------
// K1: w1 = W @ a1, w2 = W @ a2, packed into zero-padded B (FIN x 16),
//     column 0 = w1, column 1 = w2, columns 2..15 = 0.
// ---------------------------------------------------------------------------
__global__ void __launch_bounds__(256)
build_wpad(const float* __restrict__ W, const float* __restrict__ a,
           float* __restrict__ wpad) {
    int k = blockIdx.x * blockDim.x + threadIdx.x;   // 0..FIN-1
    if (k >= FIN) return;
    const float* wrow = W + (size_t)k * FOUT;
    float s1 = 0.f, s2 = 0.f;
    for (int j = 0; j < FOUT; ++j) {
        float w = wrow[j];
        s1 += w * a[j];          // a1 = a[:FOUT]
        s2 += w * a[FOUT + j];   // a2 = a[FOUT:]
    }
    float* dst = wpad + k * 16;
#pragma unroll
    for (int n = 2; n < 16; ++n) dst[n] = 0.f;
    dst[0] = s1;
    dst[1] = s2;
}

// ---------------------------------------------------------------------------
// K2: [f|g] = X(8192x512) @ wpad(512x16) via V_WMMA_F32_16X16X4_F32.
// One wave owns a 16-row tile; 128 WMMAs over K. Layouts per ISA 7.12.2:
//   A 16x4 f32 : lanes 0-15 M=lane (K pair 0,1), lanes 16-31 M=lane-16 (K 2,3)
//   B 4x16 f32 : lane%16 = N, lane/16 selects K pair, vgpr = K within pair
//   D 16x16 f32: lane%16 = N, vgpr v = M (lanes<16) / M=v+8 (lanes>=16)
// EXEC is all-ones for the whole WMMA loop (no divergence before the stores).
// ---------------------------------------------------------------------------
__global__ void __launch_bounds__(256)
fg_wmma(const float* __restrict__ input, const float* __restrict__ wpad,
        float* __restrict__ f, float* __restrict__ g) {
    const int lane  = threadIdx.x & 31;
    const int wave  = threadIdx.x >> 5;
    const int wgl   = blockIdx.x * (blockDim.x >> 5) + wave;
    const int row0  = wgl * 16;
    const int m     = lane & 15;          // M for A, N for B/D
    const int kh    = (lane >> 4) * 2;    // K sub-pair: 0 or 2
    const float* arow = input + (size_t)(row0 + m) * FIN;

    v8f c = {};
    for (int k0 = 0; k0 < FIN; k0 += 4) {
        v2f av = *(const v2f*)(arow + k0 + kh);          // A: K = k0+kh, k0+kh+1
        v2f bv;
        bv.x = wpad[(k0 + kh)     * 16 + m];             // B row K=k0+kh, col N=m
        bv.y = wpad[(k0 + kh + 1) * 16 + m];
        c = __builtin_amdgcn_wmma_f32_16x16x4_f32(
                /*neg_a=*/false, av, /*neg_b=*/false, bv,
                /*c_mod=*/(short)0, c, /*reuse_a=*/false, /*reuse_b=*/false);
    }

    // Column 0 of D = f, column 1 = g.
    const int mBase = (lane >> 4) * 8;
    if (m == 0) {
#pragma unroll
        for (int v = 0; v < 8; ++v) f[row0 + mBase + v] = c[v];
    } else if (m == 1) {
#pragma unroll
        for (int v = 0; v < 8; ++v) g[row0 + mBase + v] = c[v];
    }
}

// ---------------------------------------------------------------------------
// K3: one block per output row. e_j = lrelu(f[row] + g[j]); softmax over j.
// 256 threads x 32 elements kept in registers (exp computed exactly once).
// g (32KB) stays resident in L2 across all 8192 blocks; the 256MB output is
// streamed with non-temporal stores so it doesn't thrash L2.
// ---------------------------------------------------------------------------
__global__ void __launch_bounds__(256)
row_softmax(const float* __restrict__ f, const float* __restrict__ g,
            float* __restrict__ out) {
    const int T = 256;
    const int E = NROWS / T;              // 32 elements per thread
    const int row = blockIdx.x;
    const int t   = threadIdx.x;
    __shared__ float red[8];              // 8 waves per block

    const float fi = f[row];
    float ev[E];
    float m = -__builtin_inff();
#pragma unroll
    for (int i = 0; i < E; ++i) {
        int j = t + T * i;                // coalesced
        float x = fi + g[j];
        x = fmaxf(x, ALPHA * x);          // leaky_relu, slope 0.2
        ev[i] = x;
        m = fmaxf(m, x);
    }
    // block max reduce
#pragma unroll
    for (int off = 16; off > 0; off >>= 1)
        m = fmaxf(m, __shfl_xor(m, off, 32));
    if ((t & 31) == 0) red[t >> 5] = m;
    __syncthreads();
    float bm = red[0];
#pragma unroll
    for (int w = 1; w < 8; ++w) bm = fmaxf(bm, red[w]);
    __syncthreads();                      // before red[] reuse

    // exp + block sum reduce
    float s = 0.f;
#pragma unroll
    for (int i = 0; i < E; ++i) {
        float p = __expf(ev[i] - bm);
        ev[i] = p;
        s += p;
    }
#pragma unroll
    for (int off = 16; off > 0; off >>= 1)
        s += __shfl_xor(s, off, 32);
    if ((t & 31) == 0) red[t >> 5] = s;
    __syncthreads();
    float bs = 0.f;
#pragma unroll
    for (int w = 0; w < 8; ++w) bs += red[w];
    const float inv = 1.0f / bs;

    float* orow = out + (size_t)row * NROWS;
#pragma unroll
    for (int i = 0; i < E; ++i) {
        int j = t + T * i;
        __builtin_nontemporal_store(ev[i] * inv, orow + j);
    }
}

// ---------------------------------------------------------------------------
extern "C" void kernel_launch(void* const* d_in, const int* in_sizes, int n_in,
                              void* d_out, int out_size, void* d_ws, size_t ws_size,
                              hipStream_t stream) {
    const float* input = (const float*)d_in[0];   // (8192, 512)
    // d_in[1] = adj (8192,8192): unused by the reference output.
    const float* W     = (const float*)d_in[2];   // (512, 256)
    const float* a     = (const float*)d_in[3];   // (512, 1)
    float* out = (float*)d_out;                   // (8192, 8192)

    char* ws   = (char*)d_ws;
    float* wpad = (float*)(ws);                   // 512*16*4  = 32 KB
    float* f    = (float*)(ws + (size_t)FIN * 16 * 4);            // 32 KB
    float* g    = (float*)(ws + (size_t)FIN * 16 * 4 + NROWS * 4);// 32 KB

    build_wpad<<<(FIN + 255) / 256, 256, 0, stream>>>(W, a, wpad);

    // 8192/16 = 512 waves; 8 waves (256 threads) per block -> 64 blocks
    fg_wmma<<<NROWS / 16 / 8, 256, 0, stream>>>(input, wpad, f, g);

    row_softmax<<<NROWS, 256, 0, stream>>>(f, g, out);
}